// TransformerCRF_65309272703315
// MI455X (gfx1250) — compile-verified
//
#include <hip/hip_runtime.h>
#include <hip/hip_bf16.h>
#include <math.h>

// ---------------- problem constants ----------------
#define BATCH   64
#define SEQ     512
#define TPRIME  510          // SEQ - 2 (slice [1:-1])
#define HIDDEN  1024
#define NLAB    9
#define NPAD    16           // labels padded to 16 for WMMA N-dim
#define LEAN_TAU   0.25f
#define LEAN_DELTA 0.35f
#define O_ID    0

typedef float v2f __attribute__((ext_vector_type(2)));
typedef float v8f __attribute__((ext_vector_type(8)));

// =====================================================================
// Kernel 1: logits[b,t,:] = hidden @ emis_w^T + emis_b   (fp32 WMMA)
// Each wave computes a 16(token-rows) x 16(label, padded) tile,
// accumulating over K=1024 in steps of 4 with V_WMMA_F32_16X16X4_F32.
// Output stored N-padded (stride NPAD) into workspace.
// =====================================================================
__global__ __launch_bounds__(256)
void crf_emis_wmma(const float* __restrict__ hs,
                   const float* __restrict__ emis_w,
                   const float* __restrict__ emis_b,
                   float* __restrict__ lg /* [BATCH*SEQ][NPAD] */) {
    // emis_w^T staged in LDS: rows 0..8 = emis_w, row 9 = zeros (pad source).
    __shared__ float lw[10 * HIDDEN];          // 40 KB of 320 KB/WGP
    const int tid = threadIdx.x;
    for (int i = tid; i < 10 * HIDDEN; i += 256)
        lw[i] = (i < NLAB * HIDDEN) ? emis_w[i] : 0.0f;
    __syncthreads();

    const int wave = tid >> 5;                 // 8 waves per block
    const int lane = tid & 31;
    const int half = lane >> 4;                // 0: K={0,1}, 1: K={2,3}
    const int mrow = lane & 15;                // A-matrix row within tile
    const int tile = blockIdx.x * 8 + wave;    // 2048 tiles total
    const long rowBase = (long)tile * 16;

    const float* aRow = hs + (rowBase + mrow) * (long)HIDDEN + half * 2;
    // B operand: column n of emis_w^T == row n of emis_w; pad cols -> zero row 9
    const int n     = lane & 15;
    const int nEff  = (n < NLAB) ? n : NLAB;   // row 9 is the zero row
    const float* bRow = lw + nEff * HIDDEN + half * 2;

    v8f c = {};                                // 16x16 fp32 accumulator
#pragma unroll 4
    for (int k = 0; k < HIDDEN; k += 4) {
        v2f a = *(const v2f*)(aRow + k);       // rows striped: lane half picks K pair
        v2f b = *(const v2f*)(bRow + k);
        c = __builtin_amdgcn_wmma_f32_16x16x4_f32(
                /*neg_a=*/false, a, /*neg_b=*/false, b,
                /*c_mod=*/(short)0, c, /*reuse_a=*/false, /*reuse_b=*/false);
    }

    const float bias = (n < NLAB) ? emis_b[n] : 0.0f;
    // C/D layout: VGPR r -> M = r (lanes 0-15) / r+8 (lanes 16-31), N = lane&15
#pragma unroll
    for (int r = 0; r < 8; ++r) {
        const long row = rowBase + r + half * 8;
        lg[row * NPAD + n] = c[r] + bias;
    }
}

// =====================================================================
// Kernel 2: per-batch CRF forward (numerator + log-partition), one
// wave32 per sequence; lane j (j<9) carries alpha[j].
// =====================================================================
__device__ __forceinline__ float lane_logits_lean(const float* __restrict__ lg,
                                                  long rowOff, int lane, bool tokmask) {
    float v = (lane < NPAD) ? lg[rowOff + lane] : 0.0f;   // pad cols are exact 0
    // top-2 of the 9 real labels (pre-adjustment), first-argmax tie-break
    float m1 = -INFINITY; int a1 = 0;
#pragma unroll
    for (int i = 0; i < NLAB; ++i) {
        float vi = __shfl(v, i, 32);
        if (vi > m1) { m1 = vi; a1 = i; }
    }
    float m2 = -INFINITY;
#pragma unroll
    for (int i = 0; i < NLAB; ++i) {
        float vi = __shfl(v, i, 32);
        if (i != a1 && vi > m2) m2 = vi;
    }
    const bool uncertain = ((m1 - m2) < LEAN_TAU) && tokmask;
    if (lane == O_ID && uncertain) v += LEAN_DELTA;       // lean toward O
    return v;
}

__global__ __launch_bounds__(32)
void crf_forward(const float* __restrict__ lg,
                 const int*   __restrict__ labels,
                 const int*   __restrict__ amask,
                 const float* __restrict__ startT,
                 const float* __restrict__ endT,
                 const float* __restrict__ trans,
                 float* __restrict__ llh) {
    const int b    = blockIdx.x;
    const int lane = threadIdx.x;
    const int jEff = (lane < NLAB) ? lane : (NLAB - 1);

    // lane j holds column j of trans: tcol[i] = trans[i -> j]
    float tcol[NLAB];
#pragma unroll
    for (int i = 0; i < NLAB; ++i) tcol[i] = trans[i * NLAB + jEff];

    const long seqBase = (long)b * SEQ;

    // ---- t = 0 ----
    bool  tok0  = amask[seqBase + 1] != 0;
    float v     = lane_logits_lean(lg, (seqBase + 1) * NPAD, lane, tok0);
    int   gold  = labels[seqBase + 1];
    int   tag   = (gold == -100) ? 0 : gold;
    float score = startT[tag] + __shfl(v, tag, 32);       // numerator (uniform)
    int   last  = tag;
    int   prev  = tag;
    float alpha = ((lane < NLAB) ? startT[jEff] : 0.0f) + v;

    // ---- t = 1 .. TPRIME-1 ----
    for (int t = 1; t < TPRIME; ++t) {
        const long pos = seqBase + 1 + t;
        const bool tok = amask[pos] != 0;
        v    = lane_logits_lean(lg, pos * NPAD, lane, tok);
        gold = labels[pos];
        tag  = (gold == -100) ? 0 : gold;
        const bool mt = (gold != -100);

        // numerator (all lanes compute identical scalars)
        const float etag = __shfl(v, tag, 32);
        if (mt) { score += trans[prev * NLAB + tag] + etag; last = tag; }
        prev = tag;

        // denominator: alpha[j] <- logsumexp_i(alpha[i] + trans[i,j]) + e[j]
        float m = -INFINITY;
        float x[NLAB];
#pragma unroll
        for (int i = 0; i < NLAB; ++i) {
            x[i] = __shfl(alpha, i, 32) + tcol[i];
            m = fmaxf(m, x[i]);
        }
        float s = 0.0f;
#pragma unroll
        for (int i = 0; i < NLAB; ++i) s += __expf(x[i] - m);
        const float nxt = m + logf(s) + v;
        alpha = mt ? nxt : alpha;
    }

    // ---- finalize ----
    score += endT[last];
    float m = -INFINITY;
    float y[NLAB];
#pragma unroll
    for (int i = 0; i < NLAB; ++i) {
        y[i] = __shfl(alpha, i, 32) + endT[i];
        m = fmaxf(m, y[i]);
    }
    float s = 0.0f;
#pragma unroll
    for (int i = 0; i < NLAB; ++i) s += __expf(y[i] - m);
    const float den = m + logf(s);

    if (lane == 0) llh[b] = score - den;
}

// =====================================================================
// Kernel 3: out = -mean(llh)
// =====================================================================
__global__ void crf_reduce(const float* __restrict__ llh, float* __restrict__ out) {
    float s = 0.0f;
    for (int b = 0; b < BATCH; ++b) s += llh[b];
    out[0] = -s * (1.0f / BATCH);
}

// =====================================================================
extern "C" void kernel_launch(void* const* d_in, const int* in_sizes, int n_in,
                              void* d_out, int out_size, void* d_ws, size_t ws_size,
                              hipStream_t stream) {
    const float* hs     = (const float*)d_in[0];  // (64,512,1024)
    const float* emis_w = (const float*)d_in[1];  // (9,1024)
    const float* emis_b = (const float*)d_in[2];  // (9,)
    const float* startT = (const float*)d_in[3];  // (9,)
    const float* endT   = (const float*)d_in[4];  // (9,)
    const float* trans  = (const float*)d_in[5];  // (9,9)
    const int*   amask  = (const int*)d_in[6];    // (64,512)
    const int*   labels = (const int*)d_in[7];    // (64,512)
    float*       out    = (float*)d_out;

    float* lg  = (float*)d_ws;                    // BATCH*SEQ*NPAD floats (2 MB)
    float* llh = lg + (long)BATCH * SEQ * NPAD;   // BATCH floats

    // 2048 tiles of 16 token-rows; 8 waves/block -> 256 blocks
    crf_emis_wmma<<<256, 256, 0, stream>>>(hs, emis_w, emis_b, lg);
    crf_forward<<<BATCH, 32, 0, stream>>>(lg, labels, amask, startT, endT, trans, llh);
    crf_reduce<<<1, 1, 0, stream>>>(llh, out);
}